// Model_6571299962943
// MI455X (gfx1250) — compile-verified
//
#include <hip/hip_runtime.h>
#include <math.h>

// Problem sizes (static, from reference)
#define N_NODES 2048
#define M_EDGES 50
#define C_DIM   64
#define B_DIM   64
#define E_INC   8192
#define BC      (B_DIM * C_DIM)          // 4096
#define NEG_SLOPE 0.2f
#define MARGIN    4.2f

typedef float v2f __attribute__((ext_vector_type(2)));
typedef float v8f __attribute__((ext_vector_type(8)));

// ---------------------------------------------------------------------------
// GEMM: xw[n,b,c] = sum_k x[b,n,k] * W[k,c]   using V_WMMA_F32_16X16X4_F32
// One wave computes a 16(n) x 16(c) tile for one batch b; K-loop 16 x (K=4).
// A (16x4 f32): lanes 0-15 -> M=lane, VGPR{0,1}=K{0,1}; lanes 16-31 -> K{2,3}
// B (4x16 f32): lanes 0-15 -> N=lane, VGPR{0,1}=K{0,1}; lanes 16-31 -> K{2,3}
// D (16x16 f32): VGPR v, lanes 0-15 -> row v, col lane; lanes 16-31 -> row v+8
// ---------------------------------------------------------------------------
__global__ __launch_bounds__(256) void k_gemm(const float* __restrict__ x,
                                              const float* __restrict__ W,
                                              float* __restrict__ xw) {
  const int wave = (blockIdx.x << 3) | (threadIdx.x >> 5);
  const int lane = threadIdx.x & 31;
  const int b  = wave >> 9;            // 512 tiles (128 n-tiles * 4 c-tiles) per batch
  const int rm = wave & 511;
  const int i0 = (rm >> 2) << 4;       // n tile origin
  const int j0 = (rm & 3) << 4;        // c tile origin
  const int half = lane >> 4;
  const int r    = lane & 15;

  const float* xrow = x + ((size_t)b * N_NODES + i0 + r) * C_DIM;
  v8f acc = {};
#pragma unroll
  for (int kk = 0; kk < 16; ++kk) {
    const int kb = (kk << 2) + (half << 1);
    v2f a = *(const v2f*)(xrow + kb);                 // A frag: row i0+r, K=kb,kb+1
    v2f bb;
    bb.x = W[kb * C_DIM + j0 + r];                    // B frag: W[kb,   j0+r]
    bb.y = W[(kb + 1) * C_DIM + j0 + r];              //         W[kb+1, j0+r]
    acc = __builtin_amdgcn_wmma_f32_16x16x4_f32(false, a, false, bb,
                                                (short)0, acc, false, false);
  }
#pragma unroll
  for (int v = 0; v < 8; ++v) {
    const int n = i0 + v + (half << 3);
    xw[(size_t)n * BC + b * C_DIM + j0 + r] = acc[v];
  }
}

// ---------------------------------------------------------------------------
// Adjacency construction (deterministic, no atomics)
// ---------------------------------------------------------------------------
__global__ void k_count_nodes(const int* __restrict__ nidx, int* __restrict__ ncount) {
  const int n = blockIdx.x * blockDim.x + threadIdx.x;
  if (n >= N_NODES) return;
  int c = 0;
  for (int e = 0; e < E_INC; ++e) c += (nidx[e] == n);
  ncount[n] = c;
}
__global__ void k_count_edges(const int* __restrict__ eidx, int* __restrict__ ecount) {
  const int m = threadIdx.x;
  if (m >= M_EDGES) return;
  int c = 0;
  for (int e = 0; e < E_INC; ++e) c += (eidx[e] == m);
  ecount[m] = c;
}
__global__ void k_scan(const int* __restrict__ ncount, int* __restrict__ noffset,
                       float* __restrict__ Ddeg,
                       const int* __restrict__ ecount, int* __restrict__ eoffset,
                       float* __restrict__ Bnorm) {
  if (threadIdx.x == 0) {
    int s = 0;
    for (int n = 0; n < N_NODES; ++n) {
      noffset[n] = s; s += ncount[n]; Ddeg[n] = (float)ncount[n];
    }
    noffset[N_NODES] = s;
  } else if (threadIdx.x == 1) {
    int s = 0;
    for (int m = 0; m < M_EDGES; ++m) {
      eoffset[m] = s; s += ecount[m];
      Bnorm[m] = ecount[m] > 0 ? 1.f / (float)ecount[m] : 0.f;
    }
    eoffset[M_EDGES] = s;
  }
}
__global__ void k_fill_nodes(const int* __restrict__ nidx,
                             const int* __restrict__ noffset, int* __restrict__ nlist) {
  const int n = blockIdx.x * blockDim.x + threadIdx.x;
  if (n >= N_NODES) return;
  int wp = noffset[n];
  for (int e = 0; e < E_INC; ++e)
    if (nidx[e] == n) nlist[wp++] = e;
}
__global__ void k_fill_edges(const int* __restrict__ eidx,
                             const int* __restrict__ eoffset, int* __restrict__ elist) {
  const int m = threadIdx.x;
  if (m >= M_EDGES) return;
  int wp = eoffset[m];
  for (int e = 0; e < E_INC; ++e)
    if (eidx[e] == m) elist[wp++] = e;
}

// ---------------------------------------------------------------------------
// Per-node att dot (first half of att) + per-node total sum (for mean loss)
// thread t: b = t/4, c-chunk = (t&3)*16 .. +16
// ---------------------------------------------------------------------------
__global__ __launch_bounds__(256) void k_att_node(const float* __restrict__ xw,
                                                  const float* __restrict__ att,
                                                  float* __restrict__ a_node,
                                                  float* __restrict__ nodesum) {
  __shared__ float red[256];
  const int n = blockIdx.x, t = threadIdx.x;
  const int b = t >> 2, g = t & 3;
  const float* p = xw + (size_t)n * BC + t * 16;
  float pd = 0.f, ps = 0.f;
#pragma unroll
  for (int j = 0; j < 16; ++j) { const float v = p[j]; pd += v * att[g * 16 + j]; ps += v; }
  pd += __shfl_xor(pd, 1, 32);
  pd += __shfl_xor(pd, 2, 32);
  if (g == 0) a_node[n * B_DIM + b] = pd;
  red[t] = ps; __syncthreads();
  for (int off = 128; off; off >>= 1) { if (t < off) red[t] += red[t + off]; __syncthreads(); }
  if (t == 0) nodesum[n] = red[0];
}

// ---------------------------------------------------------------------------
// edge_sums[m,b,c] = sum_{e in edge m} xw[node(e),b,c]
// also b_edge[m,b] = <edge_sums[m,b,:], att[C:2C]>, edgesum[m] = total sum
// ---------------------------------------------------------------------------
__global__ __launch_bounds__(256) void k_edge_sums(const float* __restrict__ xw,
    const float* __restrict__ att, const int* __restrict__ nidx,
    const int* __restrict__ eoffset, const int* __restrict__ elist,
    float* __restrict__ esum, float* __restrict__ b_edge, float* __restrict__ edgesum) {
  __shared__ float red[256];
  const int m = blockIdx.x, t = threadIdx.x;
  const int b = t >> 2, g = t & 3;
  float acc[16];
#pragma unroll
  for (int j = 0; j < 16; ++j) acc[j] = 0.f;
  const int s0 = eoffset[m], s1 = eoffset[m + 1];
  for (int k = s0; k < s1; ++k) {
    const int n = nidx[elist[k]];
    const float* p = xw + (size_t)n * BC + t * 16;
#pragma unroll
    for (int j = 0; j < 16; ++j) acc[j] += p[j];
  }
  float pd = 0.f, ps = 0.f;
  float* q = esum + (size_t)m * BC + t * 16;
#pragma unroll
  for (int j = 0; j < 16; ++j) {
    q[j] = acc[j];
    pd += acc[j] * att[C_DIM + g * 16 + j];
    ps += acc[j];
  }
  pd += __shfl_xor(pd, 1, 32);
  pd += __shfl_xor(pd, 2, 32);
  if (g == 0) b_edge[m * B_DIM + b] = pd;
  red[t] = ps; __syncthreads();
  for (int off = 128; off; off >>= 1) { if (t < off) red[t] += red[t + off]; __syncthreads(); }
  if (t == 0) edgesum[m] = red[0];
}

// nrm[m,b] = safe_norm(esum[m,b,:])
__global__ void k_nrm(const float* __restrict__ esum, float* __restrict__ nrm) {
  const int id = blockIdx.x * blockDim.x + threadIdx.x;
  if (id >= M_EDGES * B_DIM) return;
  const int m = id / B_DIM, b = id % B_DIM;
  const float* p = esum + (size_t)m * BC + b * C_DIM;
  float sq = 0.f;
  for (int c = 0; c < C_DIM; ++c) sq += p[c] * p[c];
  nrm[id] = sq > 0.f ? sqrtf(sq) : 0.f;
}

// ---------------------------------------------------------------------------
// Per-node segment softmax over factored logits:
//   logit[e,b] = leaky_relu(a_node[node,b] + b_edge[edge(e),b])
// ---------------------------------------------------------------------------
__global__ __launch_bounds__(64) void k_softmax(const float* __restrict__ a_node,
    const float* __restrict__ b_edge, const int* __restrict__ eidx,
    const int* __restrict__ noffset, const int* __restrict__ nlist,
    float* __restrict__ alpha) {
  const int n = blockIdx.x, b = threadIdx.x;
  const int s0 = noffset[n], s1 = noffset[n + 1];
  if (s0 == s1) return;
  const float an = a_node[n * B_DIM + b];
  float mx = -INFINITY;
  for (int k = s0; k < s1; ++k) {
    const int m = eidx[nlist[k]];
    float l = an + b_edge[m * B_DIM + b];
    l = l > 0.f ? l : NEG_SLOPE * l;
    mx = fmaxf(mx, l);
  }
  float sum = 0.f;
  for (int k = s0; k < s1; ++k) {
    const int m = eidx[nlist[k]];
    float l = an + b_edge[m * B_DIM + b];
    l = l > 0.f ? l : NEG_SLOPE * l;
    sum += expf(l - mx);
  }
  const float inv = 1.f / (sum + 1e-16f);
  for (int k = s0; k < s1; ++k) {
    const int e = nlist[k];
    const int m = eidx[e];
    float l = an + b_edge[m * B_DIM + b];
    l = l > 0.f ? l : NEG_SLOPE * l;
    alpha[(size_t)e * B_DIM + b] = expf(l - mx) * inv;
  }
}

// Propagate pass 1: efeat[m,b,c] = Bnorm[m] * sum_{e in m} alpha[e,b]*xw[node(e),b,c]
__global__ __launch_bounds__(256) void k_edge_feat(const float* __restrict__ xw,
    const float* __restrict__ alpha, const int* __restrict__ nidx,
    const int* __restrict__ eoffset, const int* __restrict__ elist,
    const float* __restrict__ Bnorm, float* __restrict__ efeat) {
  const int m = blockIdx.x, t = threadIdx.x;
  const int b = t >> 2;
  float acc[16];
#pragma unroll
  for (int j = 0; j < 16; ++j) acc[j] = 0.f;
  const int s0 = eoffset[m], s1 = eoffset[m + 1];
  for (int k = s0; k < s1; ++k) {
    const int e = elist[k];
    const int n = nidx[e];
    const float a = alpha[(size_t)e * B_DIM + b];
    const float* p = xw + (size_t)n * BC + t * 16;
#pragma unroll
    for (int j = 0; j < 16; ++j) acc[j] += a * p[j];
  }
  const float bn = Bnorm[m];
  float* q = efeat + (size_t)m * BC + t * 16;
#pragma unroll
  for (int j = 0; j < 16; ++j) q[j] = bn * acc[j];
}

// Propagate pass 2: out[b,n,c] = D[n] * sum_{e in n} alpha[e,b]*efeat[edge(e),b,c]
__global__ __launch_bounds__(256) void k_out(const float* __restrict__ efeat,
    const float* __restrict__ alpha, const int* __restrict__ eidx,
    const int* __restrict__ noffset, const int* __restrict__ nlist,
    const float* __restrict__ Ddeg, float* __restrict__ out) {
  const int n = blockIdx.x, t = threadIdx.x;
  const int b = t >> 2, g = t & 3;
  float acc[16];
#pragma unroll
  for (int j = 0; j < 16; ++j) acc[j] = 0.f;
  const int s0 = noffset[n], s1 = noffset[n + 1];
  for (int k = s0; k < s1; ++k) {
    const int e = nlist[k];
    const int m = eidx[e];
    const float a = alpha[(size_t)e * B_DIM + b];
    const float* p = efeat + (size_t)m * BC + t * 16;
#pragma unroll
    for (int j = 0; j < 16; ++j) acc[j] += a * p[j];
  }
  const float d = Ddeg[n];
  float* q = out + (size_t)b * (N_NODES * C_DIM) + n * C_DIM + g * 16;
#pragma unroll
  for (int j = 0; j < 16; ++j) q[j] = d * acc[j];
}

// ---------------------------------------------------------------------------
// Losses
// mean(x_i - x_j) factorizes: sum_e nodesum[node(e)] - sum_e edgesum[edge(e)]
// ---------------------------------------------------------------------------
__global__ void k_init(float* __restrict__ lossbuf) {
  if (threadIdx.x < 16) lossbuf[threadIdx.x] = 0.f;
}

__global__ __launch_bounds__(256) void k_loss_mean(const float* __restrict__ nodesum,
    const float* __restrict__ edgesum, const int* __restrict__ nidx,
    const int* __restrict__ eidx, float* __restrict__ lossbuf) {
  __shared__ float r1[256], r2[256];
  const int t = threadIdx.x;
  float si = 0.f, sj = 0.f;
  for (int e = t; e < E_INC; e += 256) {
    si += nodesum[nidx[e]];
    sj += edgesum[eidx[e]];
  }
  r1[t] = si; r2[t] = sj; __syncthreads();
  for (int off = 128; off; off >>= 1) {
    if (t < off) { r1[t] += r1[t + off]; r2[t] += r2[t + off]; }
    __syncthreads();
  }
  if (t == 0) { lossbuf[0] = r1[0]; lossbuf[1] = r2[0]; }
}

__global__ __launch_bounds__(64) void k_loss_hyper(const float* __restrict__ esum,
    const float* __restrict__ nrm, float* __restrict__ lossbuf) {
  __shared__ float red[64];
  const int kk = blockIdx.x / M_EDGES, mm = blockIdx.x % M_EDGES;
  const int b = threadIdx.x;
  const float* pk = esum + (size_t)kk * BC + b * C_DIM;
  const float* pm = esum + (size_t)mm * BC + b * C_DIM;
  float inner = 0.f, d2 = 0.f;
  for (int c = 0; c < C_DIM; ++c) {
    const float vk = pk[c], vm = pm[c];
    inner += vk * vm;
    const float df = vk - vm;
    d2 += df * df;
  }
  const float dist = d2 > 0.f ? sqrtf(d2) : 0.f;
  const float cs = inner / (nrm[kk * B_DIM + b] * nrm[mm * B_DIM + b]);
  const float li = cs * dist + (1.f - cs) * fmaxf(MARGIN - dist, 0.f);
  red[b] = li; __syncthreads();
  for (int off = 32; off; off >>= 1) { if (b < off) red[b] += red[b + off]; __syncthreads(); }
  if (b == 0) atomicAdd(&lossbuf[2], fabsf(red[0] * (1.f / (float)B_DIM)));
}

__global__ void k_final(const float* __restrict__ lossbuf, float* __restrict__ out) {
  const float mean = (lossbuf[0] - lossbuf[1]) * (1.f / ((float)E_INC * (float)BC));
  out[(size_t)B_DIM * N_NODES * C_DIM] =
      fabsf(mean) + lossbuf[2] * (1.f / (float)((M_EDGES + 1) * (M_EDGES + 1)));
}

// ---------------------------------------------------------------------------
extern "C" void kernel_launch(void* const* d_in, const int* in_sizes, int n_in,
                              void* d_out, int out_size, void* d_ws, size_t ws_size,
                              hipStream_t stream) {
  (void)in_sizes; (void)n_in; (void)out_size; (void)ws_size;
  const float* x   = (const float*)d_in[0];   // [B,N,C]
  const float* W   = (const float*)d_in[1];   // [C,C]
  const float* att = (const float*)d_in[2];   // [2C]
  const int* nidx  = (const int*)d_in[3];     // [E]
  const int* eidx  = (const int*)d_in[4];     // [E]
  float* out = (float*)d_out;                 // [B,N,C] + 1 scalar

  // Workspace partition (floats then ints); ~38 MB total
  float* w = (float*)d_ws;
  float* xw      = w; w += (size_t)N_NODES * BC;   // 8,388,608
  float* esum    = w; w += (size_t)M_EDGES * BC;   // 204,800
  float* efeat   = w; w += (size_t)M_EDGES * BC;   // 204,800
  float* a_node  = w; w += N_NODES * B_DIM;        // 131,072
  float* b_edge  = w; w += M_EDGES * B_DIM;        // 3,200
  float* alpha   = w; w += (size_t)E_INC * B_DIM;  // 524,288
  float* nodesum = w; w += N_NODES;
  float* edgesum = w; w += 64;
  float* Ddeg    = w; w += N_NODES;
  float* Bnorm   = w; w += 64;
  float* nrm     = w; w += M_EDGES * B_DIM;
  float* lossbuf = w; w += 16;
  int* ip = (int*)w;
  int* ncount  = ip; ip += N_NODES;
  int* noffset = ip; ip += N_NODES + 4;
  int* nlist   = ip; ip += E_INC;
  int* ecount  = ip; ip += 64;
  int* eoffset = ip; ip += 64;
  int* elist   = ip; ip += E_INC;

  k_init<<<1, 32, 0, stream>>>(lossbuf);
  k_count_nodes<<<N_NODES / 256, 256, 0, stream>>>(nidx, ncount);
  k_count_edges<<<1, 64, 0, stream>>>(eidx, ecount);
  k_scan<<<1, 32, 0, stream>>>(ncount, noffset, Ddeg, ecount, eoffset, Bnorm);
  k_fill_nodes<<<N_NODES / 256, 256, 0, stream>>>(nidx, noffset, nlist);
  k_fill_edges<<<1, 64, 0, stream>>>(eidx, eoffset, elist);

  // GEMM: 128 n-tiles * 4 c-tiles * 64 batches = 32768 waves, 8 waves/block
  k_gemm<<<4096, 256, 0, stream>>>(x, W, xw);

  k_att_node<<<N_NODES, 256, 0, stream>>>(xw, att, a_node, nodesum);
  k_edge_sums<<<M_EDGES, 256, 0, stream>>>(xw, att, nidx, eoffset, elist,
                                           esum, b_edge, edgesum);
  k_nrm<<<(M_EDGES * B_DIM + 255) / 256, 256, 0, stream>>>(esum, nrm);
  k_softmax<<<N_NODES, 64, 0, stream>>>(a_node, b_edge, eidx, noffset, nlist, alpha);
  k_edge_feat<<<M_EDGES, 256, 0, stream>>>(xw, alpha, nidx, eoffset, elist, Bnorm, efeat);
  k_out<<<N_NODES, 256, 0, stream>>>(efeat, alpha, eidx, noffset, nlist, Ddeg, out);

  k_loss_mean<<<1, 256, 0, stream>>>(nodesum, edgesum, nidx, eidx, lossbuf);
  k_loss_hyper<<<M_EDGES * M_EDGES, 64, 0, stream>>>(esum, nrm, lossbuf);
  k_final<<<1, 1, 0, stream>>>(lossbuf, out);
}